// GraphContextPromptGenerator_83975200571522
// MI455X (gfx1250) — compile-verified
//
#include <hip/hip_runtime.h>
#include <hip/hip_bf16.h>

// ---------------------------------------------------------------------------
// Problem constants (static in the reference)
// ---------------------------------------------------------------------------
#define NN      100000          // nodes
#define EE      800000          // edges (without self loops)
#define ET      (EE + NN)       // edges + self loops
#define KGE     128
#define BB      8
#define SEQ     1024
#define HIDDEN  1024
#define DOWN    256
#define TPROMPT 32
#define PROWS   (BB * TPROMPT)  // 256 pooled rows

typedef float v2f __attribute__((ext_vector_type(2)));
typedef float v8f __attribute__((ext_vector_type(8)));

// One 16x16x4 fp32 WMMA step: D = A(16x4) * B(4x16) + C
__device__ __forceinline__ v8f wmma4(v2f a, v2f b, v8f c) {
  return __builtin_amdgcn_wmma_f32_16x16x4_f32(
      /*neg_a=*/false, a, /*neg_b=*/false, b,
      /*c_mod=*/(short)0, c, /*reuse_a=*/false, /*reuse_b=*/false);
}

// Order-preserving float <-> uint encoding for atomicMax on floats
__device__ __forceinline__ unsigned fenc(float f) {
  unsigned u = __float_as_uint(f);
  return (u & 0x80000000u) ? ~u : (u | 0x80000000u);
}
__device__ __forceinline__ float fdec(unsigned u) {
  return __uint_as_float((u & 0x80000000u) ? (u & 0x7FFFFFFFu) : ~u);
}
#define ENC_NEGINF 0x007FFFFFu   // fenc(-inf)

// ---------------------------------------------------------------------------
// 1) Embedding gather: x[i,:] = emb[graph_x[i],:]
// ---------------------------------------------------------------------------
__global__ void k_gather(const int* __restrict__ gx,
                         const float* __restrict__ emb,
                         float* __restrict__ x) {
  int idx = blockIdx.x * 256 + threadIdx.x;         // over NN*128 exactly
  int node = idx >> 7, d = idx & 127;
  x[idx] = emb[(size_t)gx[node] * KGE + d];
}

// ---------------------------------------------------------------------------
// 2a) H = X @ W  (100000x128 @ 128x128), W staged in LDS, WMMA f32 16x16x4
//     One block = one 16-row M tile, 8 waves cover the 128 columns.
// ---------------------------------------------------------------------------
__global__ void k_gemm_h(const float* __restrict__ X,
                         const float* __restrict__ W,
                         float* __restrict__ H) {
  __shared__ float sW[KGE * KGE];                   // 64 KB
  for (int i = threadIdx.x; i < KGE * KGE; i += 256) sW[i] = W[i];
  __syncthreads();

  const int wave = threadIdx.x >> 5;                // 0..7  -> N tile
  const int lane = threadIdx.x & 31;
  const int m0 = blockIdx.x * 16;
  const int n0 = wave * 16;
  const int lm = lane & 15;                         // M row / N col within tile
  const int kb = (lane >> 4) * 2;                   // K sub-block per lane half

  const float* __restrict__ arow = X + (size_t)(m0 + lm) * KGE;
  __builtin_prefetch(arow, 0, 1);

  v8f c = {};
#pragma unroll
  for (int k = 0; k < KGE; k += 4) {
    v2f a, b;
    a.x = arow[k + kb];
    a.y = arow[k + kb + 1];
    b.x = sW[(k + kb) * KGE + n0 + lm];
    b.y = sW[(k + kb + 1) * KGE + n0 + lm];
    c = wmma4(a, b, c);
  }
  const int mo = m0 + (lane >> 4) * 8;
#pragma unroll
  for (int r = 0; r < 8; ++r)
    H[(size_t)(mo + r) * KGE + n0 + lm] = c[r];
}

// ---------------------------------------------------------------------------
// 2b) Per-node attention scalars: es = h . a_src, ed = h . a_dst  (wave/node)
// ---------------------------------------------------------------------------
__global__ void k_node_scores(const float* __restrict__ H,
                              const float* __restrict__ a_s,
                              const float* __restrict__ a_d,
                              float* __restrict__ es, float* __restrict__ ed) {
  int node = blockIdx.x * 8 + (threadIdx.x >> 5);
  int lane = threadIdx.x & 31;
  const float* h = H + (size_t)node * KGE;
  float s = 0.f, d = 0.f;
#pragma unroll
  for (int j = lane; j < KGE; j += 32) { float hv = h[j]; s += hv * a_s[j]; d += hv * a_d[j]; }
#pragma unroll
  for (int off = 16; off; off >>= 1) { s += __shfl_down(s, off); d += __shfl_down(d, off); }
  if (lane == 0) { es[node] = s; ed[node] = d; }
}

// 2c) init: out rows = bias, segment max = -inf, segment sum = 0
__global__ void k_init_layer(const float* __restrict__ b,
                             float* __restrict__ O,
                             unsigned* __restrict__ nmax,
                             float* __restrict__ nsum) {
  int idx = blockIdx.x * 256 + threadIdx.x;         // NN*128 exactly
  int node = idx >> 7, d = idx & 127;
  O[idx] = b[d];
  if (d == 0) { nmax[node] = ENC_NEGINF; nsum[node] = 0.f; }
}

// 2d) pass1: e = leaky_relu(es[src]+ed[dst]); segment max via encoded atomicMax
__global__ void k_edge_pass1(const int* __restrict__ ei,
                             const float* __restrict__ es,
                             const float* __restrict__ ed,
                             float* __restrict__ ew,
                             unsigned* __restrict__ nmax) {
  int e = blockIdx.x * 256 + threadIdx.x;
  if (e >= ET) return;
  int s, d;
  if (e < EE) { s = ei[e]; d = ei[EE + e]; } else { s = d = e - EE; }
  float v = es[s] + ed[d];
  v = (v > 0.f) ? v : 0.2f * v;
  ew[e] = v;
  atomicMax(nmax + d, fenc(v));
}

// 2e) pass2: ex = exp(e - max[dst]); segment sum
__global__ void k_edge_pass2(const int* __restrict__ ei,
                             float* __restrict__ ew,
                             const unsigned* __restrict__ nmax,
                             float* __restrict__ nsum) {
  int e = blockIdx.x * 256 + threadIdx.x;
  if (e >= ET) return;
  int d = (e < EE) ? ei[EE + e] : (e - EE);
  float ex = __expf(ew[e] - fdec(nmax[d]));
  ew[e] = ex;
  atomicAdd(nsum + d, ex);
}

// 2f) aggregate: out[dst,:] += (ex/denom[dst]) * h[src,:]   (one wave / edge)
__global__ void k_edge_agg(const int* __restrict__ ei,
                           const float* __restrict__ ew,
                           const float* __restrict__ nsum,
                           const float* __restrict__ H,
                           float* __restrict__ O) {
  int e = blockIdx.x * 8 + (threadIdx.x >> 5);      // ET/8 = 112500 blocks exact
  int lane = threadIdx.x & 31;
  int s, d;
  if (e < EE) { s = ei[e]; d = ei[EE + e]; } else { s = d = e - EE; }
  float alpha = ew[e] / nsum[d];
  const float* hrow = H + (size_t)s * KGE;
  float* orow = O + (size_t)d * KGE;
  __builtin_prefetch(hrow, 0, 1);
#pragma unroll
  for (int j = lane; j < KGE; j += 32)
    atomicAdd(orow + j, alpha * hrow[j]);
}

__global__ void k_relu(float* __restrict__ O) {
  int idx = blockIdx.x * 256 + threadIdx.x;
  float v = O[idx];
  O[idx] = v > 0.f ? v : 0.f;
}

// ---------------------------------------------------------------------------
// 3) batch histogram + prefix (counts come from batch_ids)
// ---------------------------------------------------------------------------
__global__ void k_zero_counts(int* cnt) { if (threadIdx.x < 8) cnt[threadIdx.x] = 0; }
__global__ void k_hist(const int* __restrict__ bid, int* __restrict__ cnt) {
  int i = blockIdx.x * 256 + threadIdx.x;
  if (i < NN) atomicAdd(cnt + bid[i], 1);
}
__global__ void k_prefix(const int* __restrict__ cnt, int* __restrict__ offs) {
  if (threadIdx.x == 0) {
    int a = 0; offs[0] = 0;
    for (int i = 0; i < BB; ++i) { a += cnt[i]; offs[i + 1] = a; }
  }
}

// ---------------------------------------------------------------------------
// 4) Pool bins in *input* space (linearity of the down-projections):
//    per (batch,bin): SX = sum of node-feature rows in bin (128 dims),
//                     SH = sum of hidden-state rows in bin (1024 dims),
//    plus ctx-count / hs-count / bin-length.
// ---------------------------------------------------------------------------
__global__ void k_bin_sums(const float* __restrict__ Xf,
                           const float* __restrict__ hs,
                           const int* __restrict__ offs,
                           const int* __restrict__ seqlen,
                           float* __restrict__ SX,
                           float* __restrict__ SH,
                           int* __restrict__ cntb) {
  const int bin = blockIdx.x & 31;
  const int bi  = blockIdx.x >> 5;
  const int o0 = offs[bi];
  const int ni = offs[bi + 1] - o0;
  const int L  = ni + seqlen[bi];
  const int start = (bin * L) / TPROMPT;
  const int end   = ((bin + 1) * L + TPROMPT - 1) / TPROMPT;   // ceil div
  const int t = threadIdx.x;

  float sx = 0.f;
  float sh0 = 0.f, sh1 = 0.f, sh2 = 0.f, sh3 = 0.f;
  int cc = 0, ch = 0;
  for (int r = start; r < end; ++r) {
    if (r < ni) {
      if (t < KGE) sx += Xf[(size_t)(o0 + r) * KGE + t];
      ++cc;
    } else {
      const float* row = hs + ((size_t)bi * SEQ + (r - ni)) * HIDDEN;
      sh0 += row[t];  sh1 += row[t + 256];
      sh2 += row[t + 512]; sh3 += row[t + 768];
      ++ch;
    }
  }
  if (t < KGE) SX[(size_t)blockIdx.x * KGE + t] = sx;
  float* shrow = SH + (size_t)blockIdx.x * HIDDEN;
  shrow[t] = sh0; shrow[t + 256] = sh1; shrow[t + 512] = sh2; shrow[t + 768] = sh3;
  if (t == 0) {
    cntb[blockIdx.x * 4 + 0] = cc;
    cntb[blockIdx.x * 4 + 1] = ch;
    cntb[blockIdx.x * 4 + 2] = end - start;
  }
}

// ---------------------------------------------------------------------------
// 5) PRE[256,256] = silu( (SX@W_dc + SH@W_down + cc*b_dc + ch*b_down)/binlen )
//    WMMA, K = 128 (ctx space) then 1024 (hidden space) accumulated together.
// ---------------------------------------------------------------------------
__global__ void k_pool_gemm(const float* __restrict__ SX,
                            const float* __restrict__ SH,
                            const float* __restrict__ W_dc,
                            const float* __restrict__ W_down,
                            const float* __restrict__ b_dc,
                            const float* __restrict__ b_down,
                            const int* __restrict__ cntb,
                            float* __restrict__ PRE) {
  const int wave = threadIdx.x >> 5;
  const int lane = threadIdx.x & 31;
  const int m0 = blockIdx.x * 16;
  const int n0 = blockIdx.y * 128 + wave * 16;
  const int lm = lane & 15;
  const int kb = (lane >> 4) * 2;

  v8f c = {};
  const float* arow = SX + (size_t)(m0 + lm) * KGE;
#pragma unroll
  for (int k = 0; k < KGE; k += 4) {
    v2f a, b;
    a.x = arow[k + kb]; a.y = arow[k + kb + 1];
    b.x = W_dc[(k + kb) * DOWN + n0 + lm];
    b.y = W_dc[(k + kb + 1) * DOWN + n0 + lm];
    c = wmma4(a, b, c);
  }
  arow = SH + (size_t)(m0 + lm) * HIDDEN;
  for (int k = 0; k < HIDDEN; k += 4) {
    v2f a, b;
    a.x = arow[k + kb]; a.y = arow[k + kb + 1];
    b.x = W_down[(k + kb) * DOWN + n0 + lm];
    b.y = W_down[(k + kb + 1) * DOWN + n0 + lm];
    c = wmma4(a, b, c);
  }
  const int mo = m0 + (lane >> 4) * 8;
  const int nc = n0 + lm;
  const float bdc = b_dc[nc], bdn = b_down[nc];
#pragma unroll
  for (int r = 0; r < 8; ++r) {
    int row = mo + r;
    float cc = (float)cntb[row * 4 + 0];
    float ch = (float)cntb[row * 4 + 1];
    float bl = (float)cntb[row * 4 + 2];
    float v = (c[r] + cc * bdc + ch * bdn) / bl;
    v = v / (1.f + __expf(-v));                     // SiLU
    PRE[(size_t)row * DOWN + nc] = v;
  }
}

// ---------------------------------------------------------------------------
// 6) OUT[256,1024] = PRE @ W_up + b_up   (WMMA)
// ---------------------------------------------------------------------------
__global__ void k_up_gemm(const float* __restrict__ PRE,
                          const float* __restrict__ W_up,
                          const float* __restrict__ b_up,
                          float* __restrict__ OUT) {
  const int wave = threadIdx.x >> 5;
  const int lane = threadIdx.x & 31;
  const int m0 = blockIdx.x * 16;
  const int n0 = blockIdx.y * 128 + wave * 16;
  const int lm = lane & 15;
  const int kb = (lane >> 4) * 2;

  v8f c = {};
  const float* arow = PRE + (size_t)(m0 + lm) * DOWN;
#pragma unroll
  for (int k = 0; k < DOWN; k += 4) {
    v2f a, b;
    a.x = arow[k + kb]; a.y = arow[k + kb + 1];
    b.x = W_up[(k + kb) * HIDDEN + n0 + lm];
    b.y = W_up[(k + kb + 1) * HIDDEN + n0 + lm];
    c = wmma4(a, b, c);
  }
  const int mo = m0 + (lane >> 4) * 8;
  const int nc = n0 + lm;
  const float bu = b_up[nc];
#pragma unroll
  for (int r = 0; r < 8; ++r)
    OUT[(size_t)(mo + r) * HIDDEN + nc] = c[r] + bu;
}

// ---------------------------------------------------------------------------
// Host launcher
// ---------------------------------------------------------------------------
extern "C" void kernel_launch(void* const* d_in, const int* in_sizes, int n_in,
                              void* d_out, int out_size, void* d_ws, size_t ws_size,
                              hipStream_t stream) {
  const int*   graph_x    = (const int*)  d_in[0];
  const int*   edge_index = (const int*)  d_in[1];
  const int*   batch_ids  = (const int*)  d_in[2];
  const float* hidden     = (const float*)d_in[3];
  const int*   seqlen     = (const int*)  d_in[4];
  const float* emb        = (const float*)d_in[5];
  const float* gat_W      = (const float*)d_in[6];
  const float* gat_as     = (const float*)d_in[7];
  const float* gat_ad     = (const float*)d_in[8];
  const float* gat_b      = (const float*)d_in[9];
  const float* W_down     = (const float*)d_in[10];
  const float* b_down     = (const float*)d_in[11];
  const float* W_dc       = (const float*)d_in[12];
  const float* b_dc       = (const float*)d_in[13];
  const float* W_up       = (const float*)d_in[14];
  const float* b_up       = (const float*)d_in[15];
  float* out = (float*)d_out;

  // ---- workspace layout (all fp32 unless noted) ----
  const size_t FN = (size_t)NN * KGE;               // 12.8M floats per node buffer
  float* x0 = (float*)d_ws;
  float* x1 = x0 + FN;
  float* x2 = x1 + FN;
  float* es = x2 + FN;
  float* ed = es + NN;
  unsigned* nmax = (unsigned*)(ed + NN);
  float* nsum = (float*)(nmax + NN);
  float* ew   = nsum + NN;                          // ET per-edge values
  int*   cnt  = (int*)(ew + ET);                    // 8
  int*   offs = cnt + 8;                            // 9 (+pad)
  float* SX   = (float*)(offs + 16);                // 256 x 128
  float* SH   = SX + (size_t)PROWS * KGE;           // 256 x 1024
  int*   cntb = (int*)(SH + (size_t)PROWS * HIDDEN);// 256 x 4
  float* PRE  = (float*)(cntb + PROWS * 4);         // 256 x 256

  // 1) gather embeddings
  k_gather<<<(NN * KGE) / 256, 256, 0, stream>>>(graph_x, emb, x0);

  // 2) three GAT layers, rotating buffers (X dead after GEMM -> next layer's H reuses it)
  float* X = x0; float* H = x1; float* O = x2;
  for (int l = 0; l < 3; ++l) {
    k_gemm_h<<<NN / 16, 256, 0, stream>>>(X, gat_W + (size_t)l * KGE * KGE, H);
    k_node_scores<<<NN / 8, 256, 0, stream>>>(H, gat_as + l * KGE, gat_ad + l * KGE, es, ed);
    k_init_layer<<<(NN * KGE) / 256, 256, 0, stream>>>(gat_b + l * KGE, O, nmax, nsum);
    k_edge_pass1<<<(ET + 255) / 256, 256, 0, stream>>>(edge_index, es, ed, ew, nmax);
    k_edge_pass2<<<(ET + 255) / 256, 256, 0, stream>>>(edge_index, ew, nmax, nsum);
    k_edge_agg<<<ET / 8, 256, 0, stream>>>(edge_index, ew, nsum, H, O);
    if (l < 2) k_relu<<<(NN * KGE) / 256, 256, 0, stream>>>(O);
    float* nX = O; float* nH = X; float* nO = H;    // rotate
    X = nX; H = nH; O = nO;
  }
  // X now holds the final node features

  // 3) per-batch node counts / offsets
  k_zero_counts<<<1, 32, 0, stream>>>(cnt);
  k_hist<<<(NN + 255) / 256, 256, 0, stream>>>(batch_ids, cnt);
  k_prefix<<<1, 32, 0, stream>>>(cnt, offs);

  // 4) adaptive-pool bin sums in input space (exact, via linearity)
  k_bin_sums<<<PROWS, 256, 0, stream>>>(X, hidden, offs, seqlen, SX, SH, cntb);

  // 5) pooled down-projections + biases + mean + SiLU (WMMA)
  k_pool_gemm<<<dim3(PROWS / 16, 2), 256, 0, stream>>>(SX, SH, W_dc, W_down,
                                                       b_dc, b_down, cntb, PRE);

  // 6) up-projection to [8,32,1024] (WMMA)
  k_up_gemm<<<dim3(PROWS / 16, HIDDEN / 128), 256, 0, stream>>>(PRE, W_up, b_up, out);
}